// LXformer_7095285973745
// MI455X (gfx1250) — compile-verified
//
#include <hip/hip_runtime.h>
#include <hip/hip_bf16.h>

#define B0  4
#define N0  8192
#define C0  128
#define CN0 64
#define M0  32

typedef __attribute__((ext_vector_type(2))) float v2f;
typedef __attribute__((ext_vector_type(8))) float v8f;

__device__ __forceinline__ float wave_sum(float v) {
#pragma unroll
  for (int off = 16; off >= 1; off >>= 1) v += __shfl_xor(v, off, 32);
  return v;
}
__device__ __forceinline__ float wave_max(float v) {
#pragma unroll
  for (int off = 16; off >= 1; off >>= 1) v = fmaxf(v, __shfl_xor(v, off, 32));
  return v;
}

// ---------------- 1. wq1 = Wq @ w_sa1, wk1 = Wk @ w_sa1  (tiny) ----------------
__global__ void prep_k(const float* __restrict__ Wq, const float* __restrict__ Wk,
                       const float* __restrict__ w1, float* __restrict__ wq1,
                       float* __restrict__ wk1) {
  int t = threadIdx.x;  // 128
  float aq = 0.f, ak = 0.f;
#pragma unroll 8
  for (int cn = 0; cn < CN0; ++cn) {
    float w = w1[cn];
    aq += Wq[t * CN0 + cn] * w;
    ak += Wk[t * CN0 + cn] * w;
  }
  wq1[t] = aq;
  wk1[t] = ak;
}

// ---------------- 2. vsum partials: vpart[b][seg][c] = sum_n v ----------------
__global__ void vsum_part_k(const float* __restrict__ v, float* __restrict__ vpart) {
  int b = blockIdx.x, seg = blockIdx.y;  // 4 x 8
  int t = threadIdx.x;                   // 256
  int c = t & 127, half = t >> 7;
  const float* base = v + ((long)b * N0 + seg * 1024 + half) * C0 + c;
  float acc = 0.f;
  for (int i = 0; i < 512; ++i) acc += base[(long)i * (2 * C0)];
  __shared__ float red[256];
  red[t] = acc;
  __syncthreads();
  if (t < 128) vpart[((long)b * 8 + seg) * C0 + t] = red[t] + red[t + 128];
}

// ---------------- 3. qdot/kdot: one wave per row ----------------
__global__ void qk_dots_k(const float* __restrict__ q, const float* __restrict__ k,
                          const float* __restrict__ wq1, const float* __restrict__ wk1,
                          float* __restrict__ qdot, float* __restrict__ kdot) {
  int row = blockIdx.x * 8 + (threadIdx.x >> 5);
  int lane = threadIdx.x & 31;
  const float* qr = q + (long)row * C0;
  const float* kr = k + (long)row * C0;
  float aq = 0.f, ak = 0.f;
#pragma unroll
  for (int j = 0; j < 4; ++j) {
    int c = lane + 32 * j;
    aq += qr[c] * wq1[c];
    ak += kr[c] * wk1[c];
  }
  aq = wave_sum(aq);
  ak = wave_sum(ak);
  if (lane == 0) { qdot[row] = aq; kdot[row] = ak; }
}

// ---------------- 4. vp for 16 edge rows each side per batch ----------------
__global__ void vp_edge_k(const float* __restrict__ v, const float* __restrict__ Wv,
                          float* __restrict__ vedge) {
  int b = blockIdx.x, half = blockIdx.y;  // 4 x 2
  int t = threadIdx.x;                    // 256
  int cn = t & 63;
  int r0 = t >> 6;  // 0..3
  long rowg0 = (long)b * N0 + (half ? (N0 - 16) : 0);
  for (int r = r0; r < 16; r += 4) {
    const float* vr = v + (rowg0 + r) * C0;
    float acc = 0.f;
    for (int c = 0; c < C0; ++c) acc += vr[c] * Wv[c * CN0 + cn];
    vedge[(((long)b * 2 + half) * 16 + r) * CN0 + cn] = acc;
  }
}

// ---------------- 5. THE pass over pe (268 MB, read once) ----------------
// grid (B*M, 32 chunks), block 256 = 8 waves. Wave: fixed (b,m), 32 n values.
// Produces pedotT[b][m][n] and per-(bm,chunk) Spe partials.
__global__ void pe_pass_k(const float* __restrict__ pe, const float* __restrict__ w_sa1,
                          float* __restrict__ pedotT, float* __restrict__ spe_part) {
  const int bm = blockIdx.x;  // b*M + m
  const int b = bm >> 5, m = bm & 31;
  const int chunk = blockIdx.y;
  const int tid = threadIdx.x;
  const int lane = tid & 31, wid = tid >> 5;
  const int c0 = lane * 2;
  const float w0 = w_sa1[c0], w1 = w_sa1[c0 + 1];
  const int nbase = chunk * 256 + wid * 32;
  float sx = 0.f, sy = 0.f;   // Spe partial for c0, c0+1
  float myped = 0.f;          // pedot for n = nbase + lane
  const float* base = pe + (((long)b * N0 + nbase) * M0 + m) * CN0 + c0;
#pragma unroll 4
  for (int i = 0; i < 32; ++i) {
    const float2 d = *(const float2*)(base + (long)i * (M0 * CN0));
    sx += d.x;
    sy += d.y;
    float tot = wave_sum(d.x * w0 + d.y * w1);
    myped = (lane == i) ? tot : myped;
  }
  pedotT[((long)b * M0 + m) * N0 + nbase + lane] = myped;  // coalesced 128B store
  __shared__ float red[8][64];
  red[wid][c0] = sx;
  red[wid][c0 + 1] = sy;
  __syncthreads();
  if (tid < 64) {
    float s = 0.f;
#pragma unroll
    for (int w = 0; w < 8; ++w) s += red[w][tid];
    spe_part[((long)bm * 32 + chunk) * 64 + tid] = s;
  }
}

// ---------------- 6. Assemble S[b,m,c] = Spe + vtotal - edge corrections ----------------
__global__ void s_assemble_k(const float* __restrict__ vpart, const float* __restrict__ Wv,
                             const float* __restrict__ vedge, const float* __restrict__ spe_part,
                             float* __restrict__ S) {
  int b = blockIdx.x;   // 4 blocks
  int t = threadIdx.x;  // 256
  __shared__ float vsumL[C0];
  __shared__ float vtotL[CN0];
  if (t < C0) {
    float a = 0.f;
#pragma unroll
    for (int p = 0; p < 8; ++p) a += vpart[((long)b * 8 + p) * C0 + t];
    vsumL[t] = a;
  }
  __syncthreads();
  if (t < CN0) {
    float a = 0.f;
    for (int c = 0; c < C0; ++c) a += vsumL[c] * Wv[c * CN0 + t];
    vtotL[t] = a;
  }
  __syncthreads();
  for (int r = 0; r < 8; ++r) {
    int idx = t + r * 256;  // 2048 outputs per b
    int m = idx >> 6, cn = idx & 63;
    float spe = 0.f;
    const float* sp = spe_part + ((long)(b * M0 + m) * 32) * 64 + cn;
#pragma unroll 8
    for (int ch = 0; ch < 32; ++ch) spe += sp[ch * 64];
    float corr = 0.f;
    if (m >= 16) {  // window starts at m-16: subtract prefix rows 0..m-17
      const float* e0 = vedge + ((long)b * 2) * 16 * CN0 + cn;
      for (int j = 0; j <= m - 17; ++j) corr += e0[j * CN0];
    } else {        // window ends early: subtract suffix rows N+m-16..N-1
      const float* e1 = vedge + ((long)(b * 2 + 1)) * 16 * CN0 + cn;
      for (int j = m; j <= 15; ++j) corr += e1[j * CN0];
    }
    S[((long)b * M0 + m) * CN0 + cn] = vtotL[cn] - corr + spe;
  }
}

// ---------------- 7. softmax over M (one wave per n) + scores @ S ----------------
__global__ void attn_k(const float* __restrict__ qdot, const float* __restrict__ kdot,
                       const float* __restrict__ pedotT, const float* __restrict__ S,
                       float* __restrict__ pattn) {
  int row = blockIdx.x * 8 + (threadIdx.x >> 5);  // b*N + n
  int lane = threadIdx.x & 31;                    // = window index m
  int b = row >> 13, n = row & (N0 - 1);
  float qd = qdot[row];
  int idx = n + lane - 16;
  float kd = (idx >= 0 && idx < N0) ? kdot[b * N0 + idx] : 0.f;  // zero-padded k_m
  float lg = qd - kd + pedotT[((long)b * M0 + lane) * N0 + n];
  float mx = wave_max(lg);
  float p = __expf(lg - mx);
  float inv = 1.f / wave_sum(p);
  int c0 = lane, c1 = lane + 32;
  float a0 = 0.f, a1 = 0.f;
  const float* Sb = S + (long)b * M0 * CN0;
#pragma unroll
  for (int m = 0; m < 32; ++m) {
    float sm = __shfl(p, m, 32);
    a0 += sm * Sb[m * CN0 + c0];
    a1 += sm * Sb[m * CN0 + c1];
  }
  pattn[(long)row * CN0 + c0] = a0 * inv;
  pattn[(long)row * CN0 + c1] = a1 * inv;
}

// ---------------- 8. out = p_attn[32768,64] @ w_sa2[64,128] via V_WMMA_F32_16X16X4_F32 ----------------
// Block = 8 waves; wave w owns the 16x16 tile at (blockIdx.x*16, w*16). K=64 -> 16 wmma steps.
__global__ void gemm_out_k(const float* __restrict__ P, const float* __restrict__ W,
                           float* __restrict__ out) {
  const int lane = threadIdx.x & 31;
  const int wid = threadIdx.x >> 5;
  const int rbase = blockIdx.x * 16;
  const int cbase = wid * 16;
  const int lh = lane & 15;
  const int sel = lane >> 4;  // 0: K pair {k,k+1}; 1: K pair {k+2,k+3}
  v8f acc = {};
  const float* prow = P + (long)(rbase + lh) * CN0;
#pragma unroll
  for (int k = 0; k < CN0; k += 4) {
    int ka = k + sel * 2;
    v2f a, bf;
    a.x = prow[ka];
    a.y = prow[ka + 1];
    bf.x = W[ka * C0 + cbase + lh];
    bf.y = W[(ka + 1) * C0 + cbase + lh];
    acc = __builtin_amdgcn_wmma_f32_16x16x4_f32(false, a, false, bf, (short)0, acc,
                                                false, false);
  }
#pragma unroll
  for (int g = 0; g < 8; ++g) {
    int rowm = rbase + g + sel * 8;  // C/D layout: VGPR g -> M=g (lanes 0-15), M=g+8 (16-31)
    out[(long)rowm * C0 + cbase + lh] = acc[g];
  }
}

extern "C" void kernel_launch(void* const* d_in, const int* in_sizes, int n_in,
                              void* d_out, int out_size, void* d_ws, size_t ws_size,
                              hipStream_t stream) {
  const float* q     = (const float*)d_in[0];
  const float* k     = (const float*)d_in[1];
  const float* v     = (const float*)d_in[2];
  const float* pe    = (const float*)d_in[3];
  const float* Wq    = (const float*)d_in[4];
  const float* Wk    = (const float*)d_in[5];
  const float* Wv    = (const float*)d_in[6];
  const float* w_sa1 = (const float*)d_in[7];
  const float* w_sa2 = (const float*)d_in[8];
  float* out = (float*)d_out;

  float* ws = (float*)d_ws;
  float* wq1      = ws;                  // 128
  float* wk1      = wq1 + 128;           // 128
  float* qdot     = wk1 + 128;           // 32768
  float* kdot     = qdot + (B0 * N0);    // 32768
  float* vpart    = kdot + (B0 * N0);    // 4*8*128 = 4096
  float* vedge    = vpart + 4096;        // 4*2*16*64 = 8192
  float* spe_part = vedge + 8192;        // 128*32*64 = 262144
  float* pedotT   = spe_part + 262144;   // 4*32*8192 = 1048576
  float* S        = pedotT + 1048576;    // 4*32*64 = 8192
  float* pattn    = S + 8192;            // 32768*64 = 2097152

  prep_k<<<1, 128, 0, stream>>>(Wq, Wk, w_sa1, wq1, wk1);
  vsum_part_k<<<dim3(B0, 8), 256, 0, stream>>>(v, vpart);
  qk_dots_k<<<(B0 * N0) / 8, 256, 0, stream>>>(q, k, wq1, wk1, qdot, kdot);
  vp_edge_k<<<dim3(B0, 2), 256, 0, stream>>>(v, Wv, vedge);
  pe_pass_k<<<dim3(B0 * M0, 32), 256, 0, stream>>>(pe, w_sa1, pedotT, spe_part);
  s_assemble_k<<<B0, 256, 0, stream>>>(vpart, Wv, vedge, spe_part, S);
  attn_k<<<(B0 * N0) / 8, 256, 0, stream>>>(qdot, kdot, pedotT, S, pattn);
  gemm_out_k<<<(B0 * N0) / 16, 256, 0, stream>>>(pattn, w_sa2, out);
}